// LinearAttention_81295140979161
// MI455X (gfx1250) — compile-verified
//
#include <hip/hip_runtime.h>

// ---------------------------------------------------------------------------
// Linear attention (non-causal) for MI455X / gfx1250, FP32 WMMA path.
//   out[b,l,h,:] = feat(q) @ KV * z,   KV = feat(K)^T V,  z = 1/(feat(q)·ksum+eps)
// Memory-bound (256MB min traffic vs 4.3 GFLOP) -> keep FP32, use
// V_WMMA_F32_16X16X4_F32 for both GEMMs, one HBM touch per element.
// ---------------------------------------------------------------------------

typedef __attribute__((ext_vector_type(2))) float v2f;
typedef __attribute__((ext_vector_type(8))) float v8f;

#define B_        4
#define L_        4096
#define D_        1024
#define H_        16
#define HD        64            // head dim
#define NBH       (B_ * H_)     // 64 (b,h) pairs
#define SPLITS    8             // L-splits for phase-1 parallelism
#define PAIR_ELEMS (HD * HD + HD)   // 4096 KV + 64 ksum = 4160 floats
#define ROWS_PER_BLOCK 512
#define EPS       1e-6f

__device__ __forceinline__ float featf(float x) {
    // elu(x)+1 : x>0 ? x+1 : exp(x)   (branchless -> v_exp_f32 + v_cndmask)
    return x > 0.0f ? x + 1.0f : __expf(x);
}

// ---------------------------------------------------------------------------
// Phase 1: per (b,h,split) compute partial KV (64x64) and ksum (64).
// Block = 4 waves; wave i owns KV rows d = [16i, 16i+16).
// A tile = feat(K)^T (16 d-rows x 4 s-cols): lane<16 -> K={0,1}, lane>=16 -> K={2,3}
// B tile = V (4 s-rows x 16 m-cols), 4 m-slices -> 4 WMMAs per 4 sequence rows.
// ---------------------------------------------------------------------------
__global__ __launch_bounds__(128) void la_phase1(const float* __restrict__ k,
                                                 const float* __restrict__ v,
                                                 float* __restrict__ part) {
    const int split = blockIdx.x % SPLITS;
    const int bh    = blockIdx.x / SPLITS;
    const int b = bh / H_, h = bh % H_;
    const int wave = threadIdx.x >> 5;     // d-slice 0..3
    const int lane = threadIdx.x & 31;
    const int lo   = lane & 15;
    const int hi   = lane >> 4;            // 0|1 -> K pair {0,1} vs {2,3}

    const float* kb = k + (size_t)b * L_ * D_ + h * HD;
    const float* vb = v + (size_t)b * L_ * D_ + h * HD;

    v8f c0 = {}, c1 = {}, c2 = {}, c3 = {};
    float ksacc = 0.0f;

    const int sBase = split * (L_ / SPLITS);
    for (int s0 = sBase; s0 < sBase + L_ / SPLITS; s0 += 4) {
        const int sA = s0 + 2 * hi;
        // A operand: feat(k), rows d = wave*16+lo
        float a0 = featf(kb[(size_t)sA       * D_ + wave * 16 + lo]);
        float a1 = featf(kb[(size_t)(sA + 1) * D_ + wave * 16 + lo]);
        ksacc += a0 + a1;
        v2f a = {a0, a1};
        // B operand: v rows sA, sA+1, cols m = j*16+lo
        const float* vr0 = vb + (size_t)sA       * D_;
        const float* vr1 = vb + (size_t)(sA + 1) * D_;
        v2f b0 = {vr0[ 0 + lo], vr1[ 0 + lo]};
        v2f b1 = {vr0[16 + lo], vr1[16 + lo]};
        v2f b2 = {vr0[32 + lo], vr1[32 + lo]};
        v2f b3 = {vr0[48 + lo], vr1[48 + lo]};
        c0 = __builtin_amdgcn_wmma_f32_16x16x4_f32(false, a, false, b0, (short)0, c0, false, false);
        c1 = __builtin_amdgcn_wmma_f32_16x16x4_f32(false, a, false, b1, (short)0, c1, false, false);
        c2 = __builtin_amdgcn_wmma_f32_16x16x4_f32(false, a, false, b2, (short)0, c2, false, false);
        c3 = __builtin_amdgcn_wmma_f32_16x16x4_f32(false, a, false, b3, (short)0, c3, false, false);
    }

    // ksum: lanes lo and lo+16 hold complementary s-pairs (s%4 in {0,1} vs {2,3}) for the same d
    float kst = ksacc + __shfl_xor(ksacc, 16, 32);

    float* P = part + ((size_t)bh * SPLITS + split) * PAIR_ELEMS;
    // C layout: VGPR r -> (M = r + 8*hi, N = lo);  KV stored row-major [d][m]
#pragma unroll
    for (int r = 0; r < 8; ++r) {
        const int drow = wave * 16 + r + 8 * hi;
        P[drow * HD +  0 + lo] = c0[r];
        P[drow * HD + 16 + lo] = c1[r];
        P[drow * HD + 32 + lo] = c2[r];
        P[drow * HD + 48 + lo] = c3[r];
    }
    if (hi == 0) P[HD * HD + wave * 16 + lo] = kst;
}

// ---------------------------------------------------------------------------
// Deterministic split reduction: KVfinal[bh] = sum over SPLITS partials.
// ---------------------------------------------------------------------------
__global__ __launch_bounds__(256) void la_reduce(const float* __restrict__ part,
                                                 float* __restrict__ kvf) {
    const int bh = blockIdx.x;
    for (int idx = threadIdx.x; idx < PAIR_ELEMS; idx += 256) {
        float s = 0.0f;
#pragma unroll
        for (int sp = 0; sp < SPLITS; ++sp)
            s += part[((size_t)bh * SPLITS + sp) * PAIR_ELEMS + idx];
        kvf[(size_t)bh * PAIR_ELEMS + idx] = s;
    }
}

// ---------------------------------------------------------------------------
// Phase 2: out = feat(q) @ KV, scaled by z. KV + ksum staged in LDS.
// Each wave computes 16(l) x 64(m) tiles; A loaded as one global b64/lane.
// ---------------------------------------------------------------------------
__global__ __launch_bounds__(256) void la_phase2(const float* __restrict__ q,
                                                 const float* __restrict__ kvf,
                                                 float* __restrict__ out) {
    __shared__ float lds[PAIR_ELEMS];

    const int nChunks = L_ / ROWS_PER_BLOCK;       // 8
    const int chunk = blockIdx.x % nChunks;
    const int bh    = blockIdx.x / nChunks;
    const int b = bh / H_, h = bh % H_;

    for (int idx = threadIdx.x; idx < PAIR_ELEMS; idx += 256)
        lds[idx] = kvf[(size_t)bh * PAIR_ELEMS + idx];
    __syncthreads();

    const int wave = threadIdx.x >> 5;
    const int lane = threadIdx.x & 31;
    const int lo   = lane & 15;
    const int hi   = lane >> 4;

    const float* qb = q   + (size_t)b * L_ * D_ + h * HD;
    float*       ob = out + (size_t)b * L_ * D_ + h * HD;

    for (int t = wave; t < ROWS_PER_BLOCK / 16; t += 8) {
        const int l0 = chunk * ROWS_PER_BLOCK + t * 16;
        v8f c0 = {}, c1 = {}, c2 = {}, c3 = {};
        float zacc = 0.0f;
        const float* qrow = qb + (size_t)(l0 + lo) * D_;

#pragma unroll 4
        for (int d0 = 0; d0 < HD; d0 += 4) {
            const int db = d0 + 2 * hi;
            // A: one b64 per lane gives (K, K+1) in the documented layout
            float a0 = featf(qrow[db]);
            float a1 = featf(qrow[db + 1]);
            zacc += a0 * lds[HD * HD + db] + a1 * lds[HD * HD + db + 1];
            v2f a = {a0, a1};
            v2f b0 = {lds[db * HD +  0 + lo], lds[(db + 1) * HD +  0 + lo]};
            v2f b1 = {lds[db * HD + 16 + lo], lds[(db + 1) * HD + 16 + lo]};
            v2f b2 = {lds[db * HD + 32 + lo], lds[(db + 1) * HD + 32 + lo]};
            v2f b3 = {lds[db * HD + 48 + lo], lds[(db + 1) * HD + 48 + lo]};
            c0 = __builtin_amdgcn_wmma_f32_16x16x4_f32(false, a, false, b0, (short)0, c0, false, false);
            c1 = __builtin_amdgcn_wmma_f32_16x16x4_f32(false, a, false, b1, (short)0, c1, false, false);
            c2 = __builtin_amdgcn_wmma_f32_16x16x4_f32(false, a, false, b2, (short)0, c2, false, false);
            c3 = __builtin_amdgcn_wmma_f32_16x16x4_f32(false, a, false, b3, (short)0, c3, false, false);
        }

        // row dot: lanes lo / lo+16 hold complementary d-pairs (d%4 in {0,1} vs {2,3}) of row l0+lo
        float tot = zacc + __shfl_xor(zacc, 16, 32);
        float z = 1.0f / (tot + EPS);

#pragma unroll
        for (int r = 0; r < 8; ++r) {
            // accumulator VGPR r holds row M = r + 8*hi -> fetch that row's z
            float zr = __shfl(z, r + 8 * hi, 32);
            float* orow = ob + (size_t)(l0 + r + 8 * hi) * D_;
            orow[ 0 + lo] = c0[r] * zr;
            orow[16 + lo] = c1[r] * zr;
            orow[32 + lo] = c2[r] * zr;
            orow[48 + lo] = c3[r] * zr;
        }
    }
}

// ---------------------------------------------------------------------------
extern "C" void kernel_launch(void* const* d_in, const int* in_sizes, int n_in,
                              void* d_out, int out_size, void* d_ws, size_t ws_size,
                              hipStream_t stream) {
    const float* q = (const float*)d_in[0];
    const float* k = (const float*)d_in[1];
    const float* v = (const float*)d_in[2];
    float* out = (float*)d_out;

    float* part = (float*)d_ws;                                  // 64*8*4160 f32 ~8.5MB
    float* kvf  = part + (size_t)NBH * SPLITS * PAIR_ELEMS;      // 64*4160 f32 ~1.1MB

    la_phase1<<<NBH * SPLITS, 128, 0, stream>>>(k, v, part);
    la_reduce<<<NBH, 256, 0, stream>>>(part, kvf);
    la_phase2<<<NBH * (L_ / ROWS_PER_BLOCK), 256, 0, stream>>>(q, kvf, out);
}